// Easy_loss_50371376447654
// MI455X (gfx1250) — compile-verified
//
#include <hip/hip_runtime.h>
#include <stdint.h>

#define BATCH   2048
#define VOCAB   50257
#define TPB     512
#define TILE    4096
#define NTILES  ((VOCAB + TILE - 1) / TILE)   // 13
#define NB12    4096
#define CAND_MAX  2560u
#define CAND2_MAX 768u

typedef unsigned int u32x4 __attribute__((ext_vector_type(4)));
typedef int          i32x4 __attribute__((ext_vector_type(4)));
typedef int          i32x8 __attribute__((ext_vector_type(8)));

// Low 32 bits of a generic (flat) shared pointer == LDS byte offset
__device__ __forceinline__ uint32_t lds_addr_of(const void* p) {
  return (uint32_t)(uintptr_t)p;
}

// Order-preserving float -> uint key
__device__ __forceinline__ uint32_t fkey(float f) {
  uint32_t b = __float_as_uint(f);
  return (b & 0x80000000u) ? ~b : (b | 0x80000000u);
}
__device__ __forceinline__ float key_to_float(uint32_t u) {
  uint32_t b = (u & 0x80000000u) ? (u & 0x7FFFFFFFu) : ~u;
  return __uint_as_float(b);
}

// Issue one TDM 1-D tile load: TILE fp32 elements (OOB beyond `remaining` zero-fills).
// D# layout per CDNA5 ISA ch.8 (async_tensor): group0 = count/lds/global/type,
// group1 = data_size, tensor_dim0, tensor_dim1=1, tile_dim0=TILE, tile_dim1=1.
// This toolchain exposes the 6-arg builtin: (g0, g1, g2, g3, g4, cpol).
__device__ __forceinline__ void tdm_load_row_tile(const float* g, uint32_t remaining,
                                                  uint32_t lds_byte) {
  uint64_t ga = (uint64_t)(uintptr_t)g;
  u32x4 g0;
  g0.x = 1u;                                               // count=1, user mode
  g0.y = lds_byte;                                         // lds_addr [63:32]
  g0.z = (uint32_t)ga;                                     // global_addr[31:0]
  g0.w = ((uint32_t)(ga >> 32) & 0x01FFFFFFu) | (2u << 30);// global_addr[56:32], type=2
  i32x8 g1;
  g1[0] = (int)(2u << 16);                                 // data_size = 4 bytes
  g1[1] = (int)((remaining & 0xFFFFu) << 16);              // tensor_dim0[15:0]
  g1[2] = (int)(((remaining >> 16) & 0xFFFFu) | (1u << 16)); // dim0 hi | tensor_dim1=1
  g1[3] = (int)((uint32_t)TILE << 16);                     // tile_dim0 = TILE
  g1[4] = 1;                                               // tile_dim1=1, tile_dim2=0
  g1[5] = (int)VOCAB;                                      // tensor_dim0_stride lo
  g1[6] = 0;
  g1[7] = 0;
  i32x4 gz4 = (i32x4){0, 0, 0, 0};
  i32x8 gz8 = (i32x8){0, 0, 0, 0, 0, 0, 0, 0};
  __builtin_amdgcn_tensor_load_to_lds(g0, g1, gz4, gz4, gz8, 0);
}

// Stream the row through double-buffered LDS tiles staged by the TDM.
// Consumption is vectorized: ds_load_b128 (float4) per thread, scalar tail.
template <typename F>
__device__ __forceinline__ void stream_row(const float* rowp, float (*tileBuf)[TILE],
                                           F&& consume) {
  const int tid = threadIdx.x;
  if (tid == 0) tdm_load_row_tile(rowp, VOCAB, lds_addr_of(&tileBuf[0][0]));
  for (int t = 0; t < NTILES; ++t) {
    if (t + 1 < NTILES) {
      if (tid == 0)
        tdm_load_row_tile(rowp + (t + 1) * TILE, (uint32_t)(VOCAB - (t + 1) * TILE),
                          lds_addr_of(&tileBuf[(t + 1) & 1][0]));
      __builtin_amdgcn_s_wait_tensorcnt(1);   // oldest (current tile) done
    } else {
      __builtin_amdgcn_s_wait_tensorcnt(0);
    }
    __syncthreads();
    const float* tb = tileBuf[t & 1];
    const int base = t * TILE;
    const int lim = (base + TILE <= VOCAB) ? TILE : (VOCAB - base);
    const int n4 = lim >> 2;
    const float4* tb4 = reinterpret_cast<const float4*>(tb);
    for (int i = tid; i < n4; i += TPB) {
      float4 v = tb4[i];                      // ds_load_b128
      consume(v.x); consume(v.y); consume(v.z); consume(v.w);
    }
    for (int i = (n4 << 2) + tid; i < lim; i += TPB) consume(tb[i]);
    __syncthreads();                          // safe to overwrite this buffer next round
  }
}

// Top-down chunked scan of a 4096-bucket histogram: find bucket T such that
// count(buckets > T) = *outAbove < need and count(buckets >= T) >= need.
__device__ __forceinline__ void find_kth_bucket_4096(const uint32_t* h, uint32_t* chunkSum,
                                                     uint32_t need, uint32_t* outT,
                                                     uint32_t* outAbove) {
  const int tid = threadIdx.x;
  uint32_t s = 0;
#pragma unroll
  for (int j = 0; j < NB12 / TPB; ++j) s += h[tid * (NB12 / TPB) + j];
  chunkSum[tid] = s;
  __syncthreads();
  if (tid == 0) {
    uint32_t acc = 0, T = 0, above = 0;
    for (int c = TPB - 1; c >= 0; --c) {
      uint32_t cs = chunkSum[c];
      if (acc + cs >= need) {
        for (int b = c * (NB12 / TPB) + (NB12 / TPB) - 1; b >= c * (NB12 / TPB); --b) {
          uint32_t hb = h[b];
          if (acc + hb >= need) { T = (uint32_t)b; above = acc; break; }
          acc += hb;
        }
        break;
      }
      acc += cs;
    }
    *outT = T;
    *outAbove = above;
  }
  __syncthreads();
}

__global__ __launch_bounds__(TPB) void topk_lse_kernel(const float* __restrict__ features,
                                                       const int* __restrict__ labels,
                                                       const int* __restrict__ kptr,
                                                       float* __restrict__ row_out) {
  const int row = blockIdx.x;
  const int tid = threadIdx.x;
  const uint32_t K = (uint32_t)kptr[0];
  const float* rowp = features + (size_t)row * VOCAB;

  __shared__ float    tileBuf[2][TILE];   // 32 KB, TDM staging
  __shared__ uint32_t hist[NB12];         // 16 KB
  __shared__ uint32_t chunkSum[TPB];      // 2 KB
  __shared__ uint32_t cand[CAND_MAX];     // 10 KB
  __shared__ uint32_t cand2[CAND2_MAX];   // 3 KB
  __shared__ uint32_t sT, sAbove, sT2, sAbove2, sT3, sNeed3, sCandCnt, sCand2Cnt;
  __shared__ float    sFlabel;

  for (int i = tid; i < NB12; i += TPB) hist[i] = 0;
  if (tid == 0) {
    sCandCnt = 0;
    sCand2Cnt = 0;
    sFlabel = rowp[labels[row]];
  }
  __syncthreads();

  // ---- Pass A: 12-bit-key histogram over the row (TDM-staged tiles) ----
  stream_row(rowp, tileBuf, [&](float f) {
    atomicAdd(&hist[fkey(f) >> 20], 1u);
  });

  find_kth_bucket_4096(hist, chunkSum, K, &sT, &sAbove);
  const uint32_t T = sT;

  // ---- Pass C: sum exp above threshold bucket; collect bucket-T candidates ----
  float psum = 0.f;
  stream_row(rowp, tileBuf, [&](float f) {
    uint32_t b = fkey(f) >> 20;
    if (b > T) {
      psum += __expf(f);
    } else if (b == T) {
      uint32_t idx = atomicAdd(&sCandCnt, 1u);
      if (idx < CAND_MAX) cand[idx] = fkey(f);
    }
  });

  const uint32_t need = K - sAbove;                 // take `need` from bucket T (>=1)
  const uint32_t m = sCandCnt < CAND_MAX ? sCandCnt : CAND_MAX;

  // ---- Stage 1 refinement: bits 19:8 of the key ----
  for (int i = tid; i < NB12; i += TPB) hist[i] = 0;
  __syncthreads();
  for (uint32_t i = tid; i < m; i += TPB) atomicAdd(&hist[(cand[i] >> 8) & 0xFFFu], 1u);
  __syncthreads();
  find_kth_bucket_4096(hist, chunkSum, need, &sT2, &sAbove2);
  const uint32_t T2 = sT2;
  for (uint32_t i = tid; i < m; i += TPB) {
    uint32_t u = cand[i];
    uint32_t k2 = (u >> 8) & 0xFFFu;
    if (k2 > T2) {
      psum += __expf(key_to_float(u));
    } else if (k2 == T2) {
      uint32_t idx = atomicAdd(&sCand2Cnt, 1u);
      if (idx < CAND2_MAX) cand2[idx] = u;
    }
  }
  __syncthreads();
  const uint32_t need2 = need - sAbove2;

  // ---- Stage 2 refinement: bits 7:0 (exact 32-bit k-th key) ----
  for (int i = tid; i < 256; i += TPB) hist[i] = 0;
  __syncthreads();
  const uint32_t m2 = sCand2Cnt < CAND2_MAX ? sCand2Cnt : CAND2_MAX;
  for (uint32_t i = tid; i < m2; i += TPB) atomicAdd(&hist[cand2[i] & 0xFFu], 1u);
  __syncthreads();
  if (tid == 0) {
    uint32_t acc = 0, T3 = 0;
    for (int b = 255; b >= 0; --b) {
      uint32_t hb = hist[b];
      if (acc + hb >= need2) { T3 = (uint32_t)b; break; }
      acc += hb;
    }
    sT3 = T3;
    sNeed3 = need2 - acc;   // # of exact-tie elements to include
  }
  __syncthreads();
  const uint32_t T3 = sT3;
  for (uint32_t i = tid; i < m2; i += TPB) {
    uint32_t u = cand2[i];
    if ((u & 0xFFu) > T3) psum += __expf(key_to_float(u));
  }
  if (tid == 0 && sNeed3 > 0) {
    uint32_t ut = (T << 20) | (T2 << 8) | T3;   // the exact k-th key (ties identical)
    psum += (float)sNeed3 * __expf(key_to_float(ut));
  }
  __syncthreads();

  // ---- Block reduction of exp partials (reuse hist memory as float scratch) ----
  float* red = (float*)hist;
  red[tid] = psum;
  __syncthreads();
  for (int s = TPB / 2; s > 0; s >>= 1) {
    if (tid < s) red[tid] += red[tid + s];
    __syncthreads();
  }
  if (tid == 0) row_out[row] = logf(red[0]) - sFlabel;
}

__global__ __launch_bounds__(256) void reduce_mean_kernel(const float* __restrict__ row_out,
                                                          float* __restrict__ out) {
  __shared__ float red[256];
  float s = 0.f;
  for (int i = threadIdx.x; i < BATCH; i += 256) s += row_out[i];
  red[threadIdx.x] = s;
  __syncthreads();
  for (int st = 128; st > 0; st >>= 1) {
    if ((int)threadIdx.x < st) red[threadIdx.x] += red[threadIdx.x + st];
    __syncthreads();
  }
  if (threadIdx.x == 0) out[0] = red[0] / (float)BATCH;
}

extern "C" void kernel_launch(void* const* d_in, const int* in_sizes, int n_in,
                              void* d_out, int out_size, void* d_ws, size_t ws_size,
                              hipStream_t stream) {
  const float* features = (const float*)d_in[0];
  const int* labels     = (const int*)d_in[1];
  const int* kptr       = (const int*)d_in[2];
  float* row_out        = (float*)d_ws;     // BATCH floats of scratch

  topk_lse_kernel<<<BATCH, TPB, 0, stream>>>(features, labels, kptr, row_out);
  reduce_mean_kernel<<<1, 256, 0, stream>>>(row_out, (float*)d_out);
}